// GatbertEmbeddings_35467839930966
// MI455X (gfx1250) — compile-verified
//
#include <hip/hip_runtime.h>
#include <hip/hip_bf16.h>

// Problem constants from the reference
#define B_      32
#define S_      512
#define NNZ_    16384
#define MAXN_   256
#define HID_    768
#define NROWS_  (B_ * MAXN_)   // 8192 pooled rows

typedef __attribute__((ext_vector_type(2))) float v2f;
typedef __attribute__((ext_vector_type(8))) float v8f;

// ---------------------------------------------------------------------------
// Kernel 0: zero the accumulation buffer (d_out doubles as `pooled`)
// ---------------------------------------------------------------------------
__global__ __launch_bounds__(256) void zero_f4(float4* __restrict__ p, int n4) {
    int i = blockIdx.x * 256 + threadIdx.x;
    if (i < n4) p[i] = make_float4(0.f, 0.f, 0.f, 0.f);
}

// ---------------------------------------------------------------------------
// Kernel 1: weighted segment-sum scatter.
// One 256-thread block per NNZ entry; index loads are uniform (scalar path),
// embedding-row loads are coalesced, accumulation via global_atomic_add_f32.
// padding_idx==0 rows contribute zero -> skip entirely.
// ---------------------------------------------------------------------------
__global__ __launch_bounds__(256) void pool_scatter(
    const int*   __restrict__ subword_ids,
    const int*   __restrict__ mask_batch,
    const int*   __restrict__ mask_node,
    const int*   __restrict__ mask_sub,
    const float* __restrict__ mask_values,
    const float* __restrict__ emb_table,
    float*       __restrict__ pooled) {
    const int e = blockIdx.x;
    const int b = mask_batch[e];
    const int n = mask_node[e];
    const int s = mask_sub[e];
    const float v = mask_values[e];
    const int tok = subword_ids[b * S_ + s];
    if (tok == 0) return;  // padding row is zeroed in the reference

    const float* __restrict__ src = emb_table + (size_t)tok * HID_;
    float* __restrict__ dst = pooled + ((size_t)b * MAXN_ + n) * (size_t)HID_;

    int c = threadIdx.x;
    #pragma unroll
    for (int i = 0; i < HID_ / 256; ++i, c += 256) {
        atomicAdd(dst + c, src[c] * v);
    }
}

// ---------------------------------------------------------------------------
// Kernel 2: in-place LayerNorm with WMMA-based row reductions.
// One wave32 owns 16 rows. Sum and sum-of-squares of each row are computed
// with V_WMMA_F32_16X16X4_F32 against an all-ones B matrix:
//   A fragment (16x4 f32): lanes 0-15 hold {K=k,k+1}, lanes 16-31 hold {k+2,k+3}
//   D = A x ones accumulates per-row sums, replicated across all 16 columns.
// C/D layout: VGPR r holds row r (lanes 0-15) and row r+8 (lanes 16-31),
// so a single __shfl from lane 0 / lane 16 broadcasts each row statistic.
// ---------------------------------------------------------------------------
__global__ __launch_bounds__(256) void layernorm_wmma(
    float* __restrict__ buf,           // pooled in, normalized out (in place)
    const float* __restrict__ gamma,
    const float* __restrict__ beta) {
    const int gwave   = (int)((blockIdx.x * 256u + threadIdx.x) >> 5);
    const int lane    = (int)(threadIdx.x & 31u);
    const int rowBase = gwave * 16;    // 512 waves cover 8192 rows
    const int m       = lane & 15;
    const int half    = lane >> 4;     // 0 -> K=k,k+1 ; 1 -> K=k+2,k+3

    const float* __restrict__ aPtr =
        buf + (size_t)(rowBase + m) * HID_ + (half << 1);

    v8f sumAcc = {};
    v8f sqAcc  = {};
    v2f ones;  ones.x = 1.0f; ones.y = 1.0f;

    #pragma unroll 4
    for (int k = 0; k < HID_; k += 4) {
        v2f a = *(const v2f*)(aPtr + k);
        // row-sum accumulation on the matrix pipe
        sumAcc = __builtin_amdgcn_wmma_f32_16x16x4_f32(
            false, a, false, ones, (short)0, sumAcc, false, false);
        v2f a2; a2.x = a.x * a.x; a2.y = a.y * a.y;
        sqAcc = __builtin_amdgcn_wmma_f32_16x16x4_f32(
            false, a2, false, ones, (short)0, sqAcc, false, false);
    }

    const float inv = 1.0f / (float)HID_;
    for (int r = 0; r < 16; ++r) {
        const int src = (r < 8) ? 0 : 16;          // half-wave holding row r
        const float s    = __shfl(sumAcc[r & 7], src, 32);
        const float sq   = __shfl(sqAcc[r & 7],  src, 32);
        const float mean = s * inv;
        const float var  = sq * inv - mean * mean; // population variance
        const float rstd = __frsqrt_rn(var + 1e-12f);

        float* __restrict__ row = buf + (size_t)(rowBase + r) * HID_;
        #pragma unroll 4
        for (int c = lane; c < HID_; c += 32) {
            float x = row[c];                       // read-before-write: safe
            row[c] = (x - mean) * rstd * gamma[c] + beta[c];
        }
    }
}

// ---------------------------------------------------------------------------
extern "C" void kernel_launch(void* const* d_in, const int* in_sizes, int n_in,
                              void* d_out, int out_size, void* d_ws, size_t ws_size,
                              hipStream_t stream) {
    (void)in_sizes; (void)n_in; (void)d_ws; (void)ws_size;

    const int*   subword_ids = (const int*)  d_in[0];
    const int*   mask_batch  = (const int*)  d_in[1];
    const int*   mask_node   = (const int*)  d_in[2];
    const int*   mask_sub    = (const int*)  d_in[3];
    const float* mask_values = (const float*)d_in[4];
    const float* emb_table   = (const float*)d_in[5];
    const float* gamma       = (const float*)d_in[6];
    const float* beta        = (const float*)d_in[7];
    float* out = (float*)d_out;

    // 1) zero the pooled accumulator (out_size = 32*256*768)
    const int n4 = out_size >> 2;
    zero_f4<<<(n4 + 255) / 256, 256, 0, stream>>>((float4*)out, n4);

    // 2) weighted segment-sum scatter, one block per NNZ entry
    pool_scatter<<<NNZ_, 256, 0, stream>>>(
        subword_ids, mask_batch, mask_node, mask_sub, mask_values,
        emb_table, out);

    // 3) in-place LayerNorm; 8192 rows, 16 rows per wave -> 512 waves
    const int ln_blocks = (NROWS_ / 16) * 32 / 256;   // 64 blocks of 8 waves
    layernorm_wmma<<<ln_blocks, 256, 0, stream>>>(out, gamma, beta);
}